// PillarFeatureNet_89197880803690
// MI455X (gfx1250) — compile-verified
//
#include <hip/hip_runtime.h>

// ---- problem constants (match reference) ----
#define HGRID 480
#define WGRID 480
#define ZGRID 2
#define CCH   64
#define NPTS  20000
#define BATCH 2
#define HW    (HGRID * WGRID)            // 230400
#define CHW   (CCH * HW)                 // 14745600
#define ZCHW  (ZGRID * CHW)              // 29491200
#define XPERM_ELEMS (BATCH * CCH * NPTS)               // 2,560,000
#define XMAX_ELEMS  (BATCH * ZGRID * CCH * HW)         // 58,982,400
#define TILES_PER_B (NPTS / 16)          // 1250 (N divisible by 16)
#define NTILES      (BATCH * TILES_PER_B)// 2500

typedef __attribute__((ext_vector_type(2))) float v2f;
typedef __attribute__((ext_vector_type(4))) float v4f;
typedef __attribute__((ext_vector_type(8))) float v8f;

// ---------------------------------------------------------------------------
// Kernel 1: zero the dense scatter-max grid (bandwidth-bound memset, b128).
// ---------------------------------------------------------------------------
__global__ void pillar_zero_grid(float* __restrict__ p, int n4) {
  v4f z = {0.f, 0.f, 0.f, 0.f};
  for (int i = blockIdx.x * blockDim.x + threadIdx.x; i < n4;
       i += gridDim.x * blockDim.x) {
    ((v4f*)p)[i] = z;
  }
}

// ---------------------------------------------------------------------------
// Kernel 2: fused  GEMM(16x64x4 via 4x V_WMMA_F32_16X16X4_F32) -> BN -> ReLU
//           -> x_perm store ([B,C,N]) -> scatter-max atomics into dense grid.
// One wave32 processes 16 points. 8 waves / 256-thread block.
//
// WMMA f32 16x16x4 operand layout (ISA 7.12.2):
//   A (16x4, 2 VGPR/lane): lanes 0-15 hold point M=lane, K={0,1};
//                          lanes 16-31 hold point M=lane-16, K={2,3}.
//   B (4x16, 2 VGPR/lane): mirrored: lanes 0-15 K={0,1}, lanes 16-31 K={2,3},
//                          N = lane%16  (B[k][n] = W[n][k]).
//   D (16x16, 8 VGPR):     VGPR r, lane l -> M = r + 8*(l>=16), N = l%16.
// ---------------------------------------------------------------------------
__global__ __launch_bounds__(256) void pillar_wmma_scatter(
    const float* __restrict__ feats,   // [B*N, 4]
    const int*   __restrict__ coords,  // [B*N, 3]  (x,y,z)
    const float* __restrict__ Wm,      // [64, 4]
    const float* __restrict__ gamma,
    const float* __restrict__ beta,
    const float* __restrict__ rmean,
    const float* __restrict__ rvar,
    float* __restrict__ xperm,         // [B, 64, N]
    float* __restrict__ xmax) {        // [B, Z*64, H, W] (zero-initialized)
  const int lane = threadIdx.x & 31;
  const int wave = threadIdx.x >> 5;
  const int tile = blockIdx.x * 8 + wave;      // wave-uniform
  if (tile >= NTILES) return;                  // whole wave exits -> EXEC stays ~0 or all-1s

  const int hi = lane >> 4;                    // lane group (0: lanes 0-15, 1: 16-31)
  const int ln = lane & 15;
  const int b  = (tile >= TILES_PER_B) ? 1 : 0;
  const int lpbase = (tile - b * TILES_PER_B) * 16;  // point base within batch
  const int gp = tile * 16;                    // flattened point base (b*N + lpbase)

  // --- A matrix: per-lane float2 of features[point = gp+ln][k = 2*hi .. 2*hi+1]
  v2f a = *(const v2f*)(feats + (size_t)(gp + ln) * 4 + 2 * hi);

  // --- per-point voxel offset (z*CHW + x*W + y); each lane computes point `ln`
  const int* cp = coords + (size_t)(gp + ln) * 3;
  int voff = cp[2] * CHW + cp[0] * WGRID + cp[1];

  // broadcast: sof[r] = offset of point (r + 8*hi)   (ds_bpermute, wave32)
  int sof[8];
#pragma unroll
  for (int r = 0; r < 8; ++r) sof[r] = __shfl(voff, r + 8 * hi, 32);

  float* xmaxb = xmax + (size_t)b * ZCHW;

#pragma unroll
  for (int t = 0; t < 4; ++t) {
    const int c = t * 16 + ln;                 // output channel for this lane
    // --- B matrix: W[c][k] pair, k = 2*hi .. 2*hi+1
    v2f bm = *(const v2f*)(Wm + (size_t)c * 4 + 2 * hi);

    v8f acc = {};
    acc = __builtin_amdgcn_wmma_f32_16x16x4_f32(
        /*neg_a=*/false, a, /*neg_b=*/false, bm,
        /*c_mod=*/(short)0, acc, /*reuse_a=*/false, /*reuse_b=*/false);

    // --- BatchNorm1d (inference) + ReLU
    const float scale = gamma[c] * rsqrtf(rvar[c] + 1e-3f);
    const float shift = beta[c] - rmean[c] * scale;
    float y[8];
#pragma unroll
    for (int r = 0; r < 8; ++r) y[r] = fmaxf(0.f, acc[r] * scale + shift);

    // --- x_perm[b][c][lpbase + 8*hi + r] : 8 consecutive floats -> 2x b128
    size_t pbase = (size_t)b * CCH * NPTS + (size_t)c * NPTS + lpbase + 8 * hi;
    *(v4f*)(xperm + pbase)     = (v4f){y[0], y[1], y[2], y[3]};
    *(v4f*)(xperm + pbase + 4) = (v4f){y[4], y[5], y[6], y[7]};

    // --- scatter-max into grid: out[b][z*C + c][h][w]
    // values are >= 0 and grid starts at 0.0f, so signed-int max == float max
    float* chan = xmaxb + (size_t)c * HW;
#pragma unroll
    for (int r = 0; r < 8; ++r)
      atomicMax((int*)(chan + sof[r]), __float_as_int(y[r]));
  }
}

extern "C" void kernel_launch(void* const* d_in, const int* in_sizes, int n_in,
                              void* d_out, int out_size, void* d_ws, size_t ws_size,
                              hipStream_t stream) {
  const float* feats  = (const float*)d_in[0];
  const int*   coords = (const int*)d_in[1];
  const float* Wm     = (const float*)d_in[2];
  const float* gamma  = (const float*)d_in[3];
  const float* beta   = (const float*)d_in[4];
  const float* rmean  = (const float*)d_in[5];
  const float* rvar   = (const float*)d_in[6];

  float* xperm = (float*)d_out;
  float* xmax  = xperm + XPERM_ELEMS;

  // 1) zero dense grid (236 MB -> pure store bandwidth)
  pillar_zero_grid<<<2048, 256, 0, stream>>>(xmax, XMAX_ELEMS / 4);

  // 2) fused WMMA + BN + ReLU + transpose-store + scatter-max
  const int blocks = (NTILES + 7) / 8;  // 8 waves (tiles) per 256-thread block
  pillar_wmma_scatter<<<blocks, 256, 0, stream>>>(
      feats, coords, Wm, gamma, beta, rmean, rvar, xperm, xmax);
}